// PZCell_28441273434127
// MI455X (gfx1250) — compile-verified
//
#include <hip/hip_runtime.h>

typedef float v2f __attribute__((ext_vector_type(2)));
typedef float v8f __attribute__((ext_vector_type(8)));

#define T_LEN 2048
#define BLK 16
#define NBLK (T_LEN / BLK)          // 128 serial blocks
#define WAVES_PER_CTA 8
#define CTA_THREADS 256

// One wave32 handles 16 batch rows over the whole sequence.
// Per 16-step block:  Y = H*X (4x V_WMMA_F32_16X16X4_F32)  + state rank-1 terms.
__global__ __launch_bounds__(CTA_THREADS) void pz_biquad_wmma(
    const float* __restrict__ x,
    const float* __restrict__ gain_ri,
    const float* __restrict__ poles_ri,
    const float* __restrict__ zeros_ri,
    float* __restrict__ out)
{
    __shared__ float hsh[16];                       // impulse response h[0..15]
    __shared__ float xt[WAVES_PER_CTA][16][16];     // per-wave X tile [batch][time]
    __shared__ float yt[WAVES_PER_CTA][16][16];     // per-wave Y tile [batch][time]

    const int tid  = threadIdx.x;
    const int w    = tid >> 5;      // wave within CTA
    const int lane = tid & 31;
    const int n    = lane & 15;     // batch column (also A-matrix row M)
    const int hi   = lane >> 4;     // half-wave selector

    // ---- derive real biquad coefficients from complex gain/poles/zeros ----
    const float gr  = gain_ri[0],  gi  = gain_ri[1];
    const float p0r = poles_ri[0], p0i = poles_ri[1];
    const float p1r = poles_ri[2], p1i = poles_ri[3];
    const float z0r = zeros_ri[0], z0i = zeros_ri[1];
    const float z1r = zeros_ri[2], z1i = zeros_ri[3];

    const float a1 = -(p0r + p1r);                  // Re(pole_coeffs[1])
    const float a2 = p0r * p1r - p0i * p1i;         // Re(pole_coeffs[2])
    const float zc1r = -(z0r + z1r), zc1i = -(z0i + z1i);
    const float zc2r = z0r * z1r - z0i * z1i;
    const float zc2i = z0r * z1i + z0i * z1r;
    const float b0 = gr;                            // Re(gain * 1)
    const float b1 = gr * zc1r - gi * zc1i;         // Re(gain * zero_coeffs[1])
    const float b2 = gr * zc2r - gi * zc2i;         // Re(gain * zero_coeffs[2])

    // ---- recurrence tables (fully unrolled -> stay in VGPRs) ----
    float g[16];                                    // 1/(1 + a1 z^-1 + a2 z^-2)
    g[0] = 1.f; g[1] = -a1;
#pragma unroll
    for (int k = 2; k < 16; ++k) g[k] = -a1 * g[k-1] - a2 * g[k-2];

    float hh[16], R1[16], R2[16], P[16], Q[16];
#pragma unroll
    for (int k = 0; k < 16; ++k) {
        const float gm1 = (k >= 1) ? g[k-1] : 0.f;
        const float gm2 = (k >= 2) ? g[k-2] : 0.f;
        hh[k] = b0 * g[k] + b1 * gm1 + b2 * gm2;    // full impulse response
        R1[k] = b1 * g[k] + b2 * gm1;               // weight of x[t0-1]
        R2[k] = b2 * g[k];                          // weight of x[t0-2]
    }
    P[0] = -a1; P[1] = a1 * a1 - a2;                // weight of y[t0-1]
    Q[0] = -a2; Q[1] = a1 * a2;                     // weight of y[t0-2]
#pragma unroll
    for (int k = 2; k < 16; ++k) {
        P[k] = -a1 * P[k-1] - a2 * P[k-2];
        Q[k] = -a1 * Q[k-1] - a2 * Q[k-2];
    }

    if (tid == 0) {
#pragma unroll
        for (int k = 0; k < 16; ++k) hsh[k] = hh[k];
    }

    // per-lane C-init coefficients for rows m = r + 8*hi (constant indices only)
    float cr1[8], cr2[8], cp[8], cq[8];
#pragma unroll
    for (int r = 0; r < 8; ++r) {
        cr1[r] = hi ? R1[r + 8] : R1[r];
        cr2[r] = hi ? R2[r + 8] : R2[r];
        cp[r]  = hi ? P[r + 8]  : P[r];
        cq[r]  = hi ? Q[r + 8]  : Q[r];
    }

    __syncthreads();

    // A operands (loop invariant): A[kb][m][k] = H[m][4kb+k] = h[m-4kb-k]
    // layout: lanes 0-15 M=lane K={4kb+0,4kb+1}; lanes 16-31 M=lane-16 K={4kb+2,4kb+3}
    v2f A[4];
#pragma unroll
    for (int kb = 0; kb < 4; ++kb) {
        const int i0 = n - 4 * kb - 2 * hi;
        A[kb].x = (i0 >= 0) ? hsh[i0]     : 0.f;
        A[kb].y = (i0 >= 1) ? hsh[i0 - 1] : 0.f;
    }

    const long long bb = (long long)(blockIdx.x * WAVES_PER_CTA + w) * 16;
    const int srow = lane >> 1;                     // staging: batch row
    const int scol = (lane & 1) * 8;                // staging: 8 contiguous times
    const float* xrow = x   + (bb + srow) * (long long)T_LEN;
    float*       orow = out + (bb + srow) * (long long)T_LEN;

    float xm1 = 0.f, xm2 = 0.f, ym1 = 0.f, ym2 = 0.f;   // carried filter state

    for (int blk = 0; blk < NBLK; ++blk) {
        const int t0 = blk * BLK;

        // coalesced 128b tile load -> LDS transpose staging
        const float4 xa = *(const float4*)(xrow + t0 + scol);
        const float4 xb = *(const float4*)(xrow + t0 + scol + 4);
        if (blk + 1 < NBLK)
            __builtin_prefetch(xrow + t0 + BLK + scol, 0, 0);
        *(float4*)&xt[w][srow][scol]     = xa;
        *(float4*)&xt[w][srow][scol + 4] = xb;
        __builtin_amdgcn_wave_barrier();
        asm volatile("s_wait_dscnt 0x0" ::: "memory");

        // B operands: B[kb][k][n] = X[t0+4kb+k][n]
        v2f Bv[4];
#pragma unroll
        for (int kb = 0; kb < 4; ++kb)
            Bv[kb] = *(const v2f*)&xt[w][n][4 * kb + 2 * hi];

        // C init from carried state (rank-1 terms)
        v8f acc;
#pragma unroll
        for (int r = 0; r < 8; ++r)
            acc[r] = cr1[r] * xm1 + cr2[r] * xm2 + cp[r] * ym1 + cq[r] * ym2;

        // Y = H * X + C  via four chained f32 WMMAs (K = 16)
#pragma unroll
        for (int kb = 0; kb < 4; ++kb)
            acc = __builtin_amdgcn_wmma_f32_16x16x4_f32(
                false, A[kb], false, Bv[kb], (short)0, acc, false, false);

        // carry state via lane shuffles (ds_bpermute), no LDS round-trip:
        // x[t0+14],x[t0+15] live in hi-half Bv[3]; y[t0+14],y[t0+15] in hi acc[6..7]
        xm2 = __shfl(Bv[3].x, 16 + n, 32);
        xm1 = __shfl(Bv[3].y, 16 + n, 32);
        ym2 = __shfl(acc[6],  16 + n, 32);
        ym1 = __shfl(acc[7],  16 + n, 32);

        // stage Y through LDS and store coalesced 128b
        const float4 y0 = make_float4(acc[0], acc[1], acc[2], acc[3]);
        const float4 y1 = make_float4(acc[4], acc[5], acc[6], acc[7]);
        *(float4*)&yt[w][n][8 * hi]     = y0;
        *(float4*)&yt[w][n][8 * hi + 4] = y1;
        __builtin_amdgcn_wave_barrier();
        asm volatile("s_wait_dscnt 0x0" ::: "memory");

        *(float4*)(orow + t0 + scol)     = *(const float4*)&yt[w][srow][scol];
        *(float4*)(orow + t0 + scol + 4) = *(const float4*)&yt[w][srow][scol + 4];
    }
}

extern "C" void kernel_launch(void* const* d_in, const int* in_sizes, int n_in,
                              void* d_out, int out_size, void* d_ws, size_t ws_size,
                              hipStream_t stream) {
    const float* x     = (const float*)d_in[0];   // [B, T, 1] f32
    const float* gain  = (const float*)d_in[1];   // [2]
    const float* poles = (const float*)d_in[2];   // [2,2]
    const float* zeros = (const float*)d_in[3];   // [2,2]
    float* out = (float*)d_out;                   // [B, T, 1] f32

    const int B = in_sizes[0] / T_LEN;            // 4096
    const int blocks = B / (WAVES_PER_CTA * 16);  // 16 batch rows per wave
    pz_biquad_wmma<<<blocks, CTA_THREADS, 0, stream>>>(x, gain, poles, zeros, out);
}